// JPEGDecoder_87368224735248
// MI455X (gfx1250) — compile-verified
//
#include <hip/hip_runtime.h>
#include <hip/hip_bf16.h>

// ---------------------------------------------------------------------------
// Fused JPEG decoder for MI455X (gfx1250, wave32).
// One wave handles 4 consecutive 8x8 blocks (all 3 channels):
//   load -> predictor (shfl-reduced dot) -> qf/scale -> dequant ->
//   IDCT via v_wmma_f32_16x16x4_f32 (2 stages, LDS transpose between) ->
//   YCbCr->RGB + qf/scale maps.
// ---------------------------------------------------------------------------

typedef __attribute__((ext_vector_type(2))) float v2f;
typedef __attribute__((ext_vector_type(8))) float v8f;

#define B_N   8
#define HB_N  128
#define WB_N  128
#define HPIX  1024
#define WPIX  1024

__device__ __constant__ float YQ[64] = {
  16,11,10,16,24,40,51,61,  12,12,14,19,26,58,60,55,
  14,13,16,24,40,57,69,56,  14,17,22,29,51,87,80,62,
  18,22,37,56,68,109,103,77, 24,36,55,64,81,104,113,92,
  49,64,78,87,103,121,120,101, 72,92,95,98,112,100,103,99};
__device__ __constant__ float CQ[64] = {
  17,18,24,47,99,99,99,99,  18,21,26,66,99,99,99,99,
  24,26,56,99,99,99,99,99,  47,66,99,99,99,99,99,99,
  99,99,99,99,99,99,99,99,  99,99,99,99,99,99,99,99,
  99,99,99,99,99,99,99,99,  99,99,99,99,99,99,99,99};

// Orthonormal DCT-II matrix element C[r][c] (r = frequency, c = sample).
__device__ inline float dct_c(int r, int c) {
  float v = __cosf(0.19634954084936207f * (float)((2 * c + 1) * r)) * 0.5f;
  return (r == 0) ? v * 0.70710678118654752f : v;
}

// D(16x16,f32) = A(16x4,f32) * B(4x16,f32) + C   -- CDNA5 WMMA
__device__ inline v8f wmma4(v2f a, v2f b, v8f c) {
  return __builtin_amdgcn_wmma_f32_16x16x4_f32(
      /*neg_a=*/false, a, /*neg_b=*/false, b,
      /*c_mod=*/(short)0, c, /*reuse_a=*/false, /*reuse_b=*/false);
}

__device__ inline float qf_to_scale(float qf) {
  float q = fmaxf(qf, 1.0f);
  float s = (q < 50.0f) ? (5000.0f / q) : (200.0f - 2.0f * q);
  return rintf(s) * 0.01f;
}

__device__ inline float sigm(float z) { return 1.0f / (1.0f + __expf(-z)); }

__global__ __launch_bounds__(256) void jpeg_dec_kernel(
    const float* __restrict__ qdct, const float* __restrict__ Wp,
    const float* __restrict__ bp,
    const int* __restrict__ pymin, const int* __restrict__ pymax,
    const int* __restrict__ pcmin, const int* __restrict__ pcmax,
    float* __restrict__ out)
{
  // per-wave 16x16 transpose buffer, pitch 18 floats (conflict/alignment pad)
  __shared__ float lds[8][16 * 18];

  const int tid  = threadIdx.x;
  const int wv   = tid >> 5;
  const int lane = tid & 31;

  const int task = blockIdx.x * 8 + wv;     // 32768 total = 8 * 128 * 32
  const int b    = task >> 12;              // / (128*32)
  const int rem  = task & 4095;
  const int hb   = rem >> 5;
  const int wq   = rem & 31;
  const int wb0  = wq << 2;                 // first of 4 blocks along Wb

  const int n   = lane & 15;                // matrix row (B) / M (A)
  const int h   = lane >> 4;                // lane-half selects K parity group
  const int u   = n & 7;                    // DCT row within block
  const int top = n >> 3;                   // 0: blocks {0,1}, 1: blocks {2,3}
  const int v0  = 2 * h;                    // first column pair
  const int v1  = 4 + 2 * h;                // second column pair

  // ---- A operand: diag2(C^T), sliced 16x4 along K ------------------------
  // A layout: lanes 0-15 hold K=4s+0,4s+1 ; lanes 16-31 hold K=4s+2,4s+3.
  v2f A[4];
#pragma unroll
  for (int s = 0; s < 4; ++s) {
#pragma unroll
    for (int j = 0; j < 2; ++j) {
      int k = 4 * s + 2 * h + j;
      float val = 0.0f;
      if ((n < 8) == (k < 8)) val = dct_c(k & 7, n & 7);  // diag2(C^T)[n][k]
      A[s][j] = val;
    }
  }

  // ---- Load raw coefficients (B operand of stage A = Xpack^T) ------------
  // Lane holds Xpack[n][k] for k in {2h,2h+1, 4+2h,4+2h+1, 8+.., 12+..}:
  // slices 0,1 from "left" block, 2,3 from "right" block; contiguous float2s.
  float2 xv[3][4];
  float2 accL = make_float2(0.f, 0.f), accR = make_float2(0.f, 0.f);
  const float2* W2 = (const float2*)Wp;     // W2[f] = (W[f][0], W[f][1])
#pragma unroll
  for (int ch = 0; ch < 3; ++ch) {
    size_t base = ((((size_t)b * 3 + ch) * HB_N + hb) * WB_N + (wb0 + top * 2)) * 64
                  + (size_t)u * 8;
    float2 xL0 = *(const float2*)(qdct + base + v0);
    float2 xL1 = *(const float2*)(qdct + base + v1);
    float2 xR0 = *(const float2*)(qdct + base + 64 + v0);
    float2 xR1 = *(const float2*)(qdct + base + 64 + v1);
    xv[ch][0] = xL0; xv[ch][1] = xL1; xv[ch][2] = xR0; xv[ch][3] = xR1;

    // predictor partials: feats index f = u*24 + v*3 + ch
    float2 wa = W2[u * 24 + (v0    ) * 3 + ch];
    float2 wb = W2[u * 24 + (v0 + 1) * 3 + ch];
    float2 wc = W2[u * 24 + (v1    ) * 3 + ch];
    float2 wd = W2[u * 24 + (v1 + 1) * 3 + ch];
    accL.x += xL0.x * wa.x + xL0.y * wb.x + xL1.x * wc.x + xL1.y * wd.x;
    accL.y += xL0.x * wa.y + xL0.y * wb.y + xL1.x * wc.y + xL1.y * wd.y;
    accR.x += xR0.x * wa.x + xR0.y * wb.x + xR1.x * wc.x + xR1.y * wd.x;
    accR.y += xR0.x * wa.y + xR0.y * wb.y + xR1.x * wc.y + xR1.y * wd.y;
  }

  // ---- Reduce partial dots over the 16-lane block groups -----------------
  // masks {1,2,4} reduce u; mask 16 reduces the v-parity half (h).
  {
    const int masks[4] = {1, 2, 4, 16};
#pragma unroll
    for (int i = 0; i < 4; ++i) {
      int mm = masks[i];
      accL.x += __shfl_xor(accL.x, mm, 32);
      accL.y += __shfl_xor(accL.y, mm, 32);
      accR.x += __shfl_xor(accR.x, mm, 32);
      accR.y += __shfl_xor(accR.y, mm, 32);
    }
  }

  // ---- rank -> qf -> scale (round-half-even like jnp.round) --------------
  const float ylo = (float)pymin[0], yhi = (float)pymax[0];
  const float clo = (float)pcmin[0], chi = (float)pcmax[0];
  const float b0 = bp[0], b1 = bp[1];
  float qfL_y = rintf(ylo + sigm(accL.x + b0) * (yhi - ylo));
  float qfL_c = rintf(clo + sigm(accL.y + b1) * (chi - clo));
  float qfR_y = rintf(ylo + sigm(accR.x + b0) * (yhi - ylo));
  float qfR_c = rintf(clo + sigm(accR.y + b1) * (chi - clo));
  float sL_y = qf_to_scale(qfL_y), sL_c = qf_to_scale(qfL_c);
  float sR_y = qf_to_scale(qfR_y), sR_c = qf_to_scale(qfR_c);

  // ---- per-lane quant-table entries --------------------------------------
  float2 qy0 = make_float2(YQ[u * 8 + v0], YQ[u * 8 + v0 + 1]);
  float2 qy1 = make_float2(YQ[u * 8 + v1], YQ[u * 8 + v1 + 1]);
  float2 qc0 = make_float2(CQ[u * 8 + v0], CQ[u * 8 + v0 + 1]);
  float2 qc1 = make_float2(CQ[u * 8 + v1], CQ[u * 8 + v1 + 1]);

  // ---- dequant + two-stage WMMA IDCT per channel -------------------------
  v8f pix[3];
  float* T = lds[wv];
#pragma unroll
  for (int ch = 0; ch < 3; ++ch) {
    float  sL = (ch == 0) ? sL_y : sL_c;
    float  sR = (ch == 0) ? sR_y : sR_c;
    float2 t0 = (ch == 0) ? qy0 : qc0;
    float2 t1 = (ch == 0) ? qy1 : qc1;

    v2f Bs[4];
    Bs[0][0] = xv[ch][0].x * (sL * t0.x); Bs[0][1] = xv[ch][0].y * (sL * t0.y);
    Bs[1][0] = xv[ch][1].x * (sL * t1.x); Bs[1][1] = xv[ch][1].y * (sL * t1.y);
    Bs[2][0] = xv[ch][2].x * (sR * t0.x); Bs[2][1] = xv[ch][2].y * (sR * t0.y);
    Bs[3][0] = xv[ch][3].x * (sR * t1.x); Bs[3][1] = xv[ch][3].y * (sR * t1.y);

    // Stage A: D1 = diag2(C^T) * Xpack^T   (quadrants = (Xq C)^T)
    v8f d1 = {0.f, 0.f, 0.f, 0.f, 0.f, 0.f, 0.f, 0.f};
#pragma unroll
    for (int s = 0; s < 4; ++s) d1 = wmma4(A[s], Bs[s], d1);

    // In-wave LDS transpose: stage-B B-operand is exactly D1^T.
    // D layout: vgpr r, lane(n,h) holds D1[r+8h][n]  -> store row-major pitch 18.
#pragma unroll
    for (int r = 0; r < 8; ++r) T[(r + 8 * h) * 18 + n] = d1[r];
    // read D1[n][k] for k = 4s+2h+{0,1}  (float2, even offsets)
    v2f Bb[4];
#pragma unroll
    for (int s = 0; s < 4; ++s) {
      float2 t = *(const float2*)&T[n * 18 + 4 * s + 2 * h];
      Bb[s][0] = t.x; Bb[s][1] = t.y;
    }

    // Stage B: D2 = diag2(C^T) * D1^T  -> spatial pixels, lane L = column x=L
    v8f o = {0.f, 0.f, 0.f, 0.f, 0.f, 0.f, 0.f, 0.f};
#pragma unroll
    for (int s = 0; s < 4; ++s) o = wmma4(A[s], Bb[s], o);
    pix[ch] = o;
  }

  // ---- redistribute qf/scale to the pixel-block owned by this lane -------
  // lane L owns x_local = L, pixel block = L/8 = 2*h + bit3(n_src-needed)
  int  src  = (lane & 7) | (((lane >> 4) & 1) << 3);
  bool selR = ((lane >> 3) & 1) != 0;
  float qy = selR ? __shfl(qfR_y, src, 32) : __shfl(qfL_y, src, 32);
  float qc = selR ? __shfl(qfR_c, src, 32) : __shfl(qfL_c, src, 32);
  float sy = selR ? __shfl(sR_y, src, 32) : __shfl(sL_y, src, 32);
  float sc = selR ? __shfl(sR_c, src, 32) : __shfl(sL_c, src, 32);

  // ---- color convert + store ---------------------------------------------
  const size_t QF_OFF = (size_t)B_N * HPIX * WPIX * 3;             // 25165824
  const size_t SC_OFF = QF_OFF + (size_t)B_N * HPIX * WPIX * 2;    // 41943040
  const int x = wq * 32 + lane;
  float2* qf_map = (float2*)(out + QF_OFF);
  float2* sc_map = (float2*)(out + SC_OFF);
#pragma unroll
  for (int i = 0; i < 8; ++i) {
    int y = hb * 8 + i;
    size_t p = (((size_t)b * HPIX) + y) * WPIX + x;
    float Y  = pix[0][i] + 128.0f;
    float Cb = pix[1][i];               // (+128 then -128)
    float Cr = pix[2][i];
    float rr = Y + 1.402f * Cr;
    float gg = Y - 0.344136f * Cb - 0.714136f * Cr;
    float bb = Y + 1.772f * Cb;
    rr = fminf(fmaxf(rintf(rr), 0.0f), 255.0f);
    gg = fminf(fmaxf(rintf(gg), 0.0f), 255.0f);
    bb = fminf(fmaxf(rintf(bb), 0.0f), 255.0f);
    out[p * 3 + 0] = rr;
    out[p * 3 + 1] = gg;
    out[p * 3 + 2] = bb;
    qf_map[p] = make_float2(qy, qc);
    sc_map[p] = make_float2(sy, sc);
  }
}

extern "C" void kernel_launch(void* const* d_in, const int* in_sizes, int n_in,
                              void* d_out, int out_size, void* d_ws, size_t ws_size,
                              hipStream_t stream) {
  const float* qdct = (const float*)d_in[0];
  const float* Wp   = (const float*)d_in[1];
  const float* bp   = (const float*)d_in[2];
  const int*   ymin = (const int*)d_in[3];
  const int*   ymax = (const int*)d_in[4];
  const int*   cmin = (const int*)d_in[5];
  const int*   cmax = (const int*)d_in[6];
  float* out = (float*)d_out;

  // 32768 waves (one per 4-block group), 8 waves per 256-thread workgroup.
  dim3 grid(4096), block(256);
  hipLaunchKernelGGL(jpeg_dec_kernel, grid, block, 0, stream,
                     qdct, Wp, bp, ymin, ymax, cmin, cmax, out);
}